// BinaryPositionEmbedding_13194139533906
// MI455X (gfx1250) — compile-verified
//
#include <hip/hip_runtime.h>

typedef __attribute__((ext_vector_type(2))) float v2f;
typedef __attribute__((ext_vector_type(8))) float v8f;

#define N_BITS  13
#define D_MODEL 1024
// M = 4*8192 = 32768 rows -> 2048 M-tiles of 16; N = 1024 -> 64 N-tiles of 16.

// Block: 256 threads = 8 waves. Grid: (64, 32).
// Wave w of block (bx, by) computes N-tile bx, M-tiles (by*8 + w)*8 .. +7.
// B (emb) regs depend only on the N-tile -> loaded once per wave.
// A (mask bits) synthesized in VGPRs from x; bits >= 13 are naturally zero.
__global__ __launch_bounds__(256)
void bpe_wmma_f32_kernel(const int* __restrict__ x,
                         const float* __restrict__ emb,
                         float* __restrict__ out) {
    const int lane = threadIdx.x & 31;
    const int wave = threadIdx.x >> 5;
    const int half = lane >> 4;    // 0 for lanes 0-15, 1 for lanes 16-31
    const int l16  = lane & 15;

    const int tile_n = blockIdx.x;           // 0..63
    const int col    = tile_n * 16 + l16;    // output column this lane owns

    // B operands for the four K-groups (K padded 13 -> 16).
    // 32-bit B 4x16 layout (mirrors documented A 16x4): lanes 0-15 carry rows
    // kbase+0/kbase+1 in VGPR0/VGPR1, lanes 16-31 carry rows kbase+2/kbase+3.
    v2f b[4];
#pragma unroll
    for (int g = 0; g < 4; ++g) {
        const int k0 = 4 * g + 2 * half;
        const int k1 = k0 + 1;
        b[g].x = (k0 < N_BITS) ? emb[k0 * D_MODEL + col] : 0.0f;
        b[g].y = (k1 < N_BITS) ? emb[k1 * D_MODEL + col] : 0.0f;
    }

    const int tile_m_base = (blockIdx.y * 8 + wave) * 8;

#pragma unroll 2
    for (int i = 0; i < 8; ++i) {
        const int tile_m = tile_m_base + i;
        const int row    = tile_m * 16 + l16;        // A row for this lane
        const unsigned xv = (unsigned)x[row];

        v8f c = {};
#pragma unroll
        for (int g = 0; g < 4; ++g) {
            const int k0 = 4 * g + 2 * half;
            v2f a;
            // mask bits as 0.0/1.0 fp32; x < 2^13 so k >= 13 gives 0 for free
            a.x = (float)((xv >> k0) & 1u);
            a.y = (float)((xv >> (k0 + 1)) & 1u);
            // D = A(16x4) * B(4x16) + C, full fp32
            c = __builtin_amdgcn_wmma_f32_16x16x4_f32(
                    /*neg_a=*/false, a, /*neg_b=*/false, b[g],
                    /*c_mod=*/(short)0, c, /*reuse_a=*/false, /*reuse_b=*/false);
        }

        // D layout: VGPR r -> row (r + 8*half), col = lane&15.
        float* o = out + (size_t)(tile_m * 16 + half * 8) * D_MODEL + col;
#pragma unroll
        for (int r = 0; r < 8; ++r)
            o[(size_t)r * D_MODEL] = c[r];
    }
}

extern "C" void kernel_launch(void* const* d_in, const int* in_sizes, int n_in,
                              void* d_out, int out_size, void* d_ws, size_t ws_size,
                              hipStream_t stream) {
    const int*   x   = (const int*)d_in[0];    // (4, 8192) int32
    const float* emb = (const float*)d_in[1];  // (13, 1024) fp32
    float*       out = (float*)d_out;          // (4, 8192, 1024) fp32

    dim3 grid(64, 32);   // 64 N-tiles x (32*8 waves covering 2048 M-tiles)
    dim3 block(256);     // 8 waves of 32
    hipLaunchKernelGGL(bpe_wmma_f32_kernel, grid, block, 0, stream, x, emb, out);
}